// PrototypicalLoss_86122684219742
// MI455X (gfx1250) — compile-verified
//
#include <hip/hip_runtime.h>
#include <hip/hip_bf16.h>

// Problem constants (fixed by setup_inputs()).
#define N_TOTAL 32768
#define D       128
#define NC      64
#define M       16384          // support rows = x[0:M], target rows = x[M:2M]
#define EPSF    1e-5f
#define DPAD    136            // 128 + 8 floats: keeps float4 (16B) alignment,
                               // breaks the 64-bank stride-128 conflict pattern

typedef __attribute__((ext_vector_type(2))) float v2f;
typedef __attribute__((ext_vector_type(8))) float v8f;

// ---------------------------------------------------------------------------
// Kernel 1: class histogram (counts + empty-class fixup). Tiny: one block.
// ---------------------------------------------------------------------------
__global__ void __launch_bounds__(256) k_counts(const int* __restrict__ y,
                                                float* __restrict__ counts) {
    __shared__ int hist[NC];
    int t = threadIdx.x;
    if (t < NC) hist[t] = 0;
    __syncthreads();
    for (int i = t; i < M; i += 256) atomicAdd(&hist[y[i]], 1);
    __syncthreads();
    if (t < NC) {
        float c = (float)hist[t];
        counts[t] = c + (c < 0.01f ? 1.0f : 0.0f);   // matches reference fixup
    }
}

// ---------------------------------------------------------------------------
// Kernel 2: prototype sums as a one-hot GEMM on the f32 matrix core.
//   P_sum[c,d] = sum_k onehot(y[k]==c) * x[k,d]
// D = A(16x4 one-hot) x B(4x16 slice of x) + C, K swept over 16384 in steps
// of 4.  32 output tiles (4 class-tiles x 8 d-tiles), one wave32 per tile.
// Deterministic, atomic-free; 268 MFLOP -> microseconds on the matrix core.
// ---------------------------------------------------------------------------
__global__ void __launch_bounds__(32) k_proto(const float* __restrict__ x,
                                              const int*   __restrict__ y,
                                              float* __restrict__ proto_sum) {
    const int tile  = blockIdx.x;      // 0..31
    const int cbase = (tile >> 3) * 16;
    const int dbase = (tile & 7) * 16;
    const int lane  = threadIdx.x;     // wave32
    const int half  = lane >> 4;       // 0: lanes 0-15, 1: lanes 16-31
    const int lm    = lane & 15;

    v8f acc = {};
    #pragma unroll 2
    for (int k0 = 0; k0 < M; k0 += 4) {
        // Wave-uniform label quad -> scalar load; select this half's K pair.
        const int4 yq = *(const int4*)(y + k0);
        const int  y0 = half ? yq.z : yq.x;      // K = 2*half
        const int  y1 = half ? yq.w : yq.y;      // K = 2*half + 1
        // A (16x4 one-hot, f32): VGPR v holds K = v + 2*half, M = lm.
        v2f a;
        a.x = (y0 == cbase + lm) ? 1.0f : 0.0f;
        a.y = (y1 == cbase + lm) ? 1.0f : 0.0f;
        // B (4x16 slice of x_support): VGPR v holds K = v + 2*half, N = lm.
        const size_t r = (size_t)(k0 + 2 * half) * D + dbase + lm;
        v2f b;
        b.x = x[r];
        b.y = x[r + D];
        // (neg_a, A, neg_b, B, c_mod, C, reuse_a, reuse_b)
        acc = __builtin_amdgcn_wmma_f32_16x16x4_f32(
                false, a, false, b, (short)0, acc, false, false);
    }
    // C layout: VGPR v -> class cbase + v + 8*half, column dbase + lm.
    #pragma unroll
    for (int v = 0; v < 8; ++v) {
        proto_sum[(size_t)(cbase + v + 8 * half) * D + dbase + lm] = acc[v];
    }
}

// ---------------------------------------------------------------------------
// Kernel 3: distances + log-softmax + per-block loss partials.
// sign(diff)*(diff+eps)^2 is not bilinear -> pure f32 VALU with prototypes
// staged (pre-divided) in LDS.  64 blocks x 256 threads; each wave32 owns 32
// target rows; within a row, lane L handles classes L and L+32; wave-level
// shfl_xor reductions implement the 64-wide softmax.
// ---------------------------------------------------------------------------
__device__ __forceinline__ float signed_sq(float diff) {
    float t  = diff + EPSF;
    float sq = t * t;
    return diff > 0.0f ? sq : (diff < 0.0f ? -sq : 0.0f);
}

__global__ void __launch_bounds__(256) k_loss(const float* __restrict__ x,
                                              const int*   __restrict__ y,
                                              const float* __restrict__ proto_sum,
                                              const float* __restrict__ counts,
                                              float* __restrict__ partials) {
    __shared__ float sp[NC * DPAD];   // prototypes (already divided by counts)
    __shared__ float wsum[8];

    const int tid = threadIdx.x;
    for (int idx = tid; idx < NC * D; idx += 256) {
        int c = idx >> 7;             // / 128
        int d = idx & (D - 1);
        sp[c * DPAD + d] = proto_sum[idx] / counts[c];
    }
    __syncthreads();

    const int wave    = tid >> 5;
    const int lane    = tid & 31;
    const int rowbase = blockIdx.x * 256 + wave * 32;

    const float4* p0 = (const float4*)(sp + (size_t)lane * DPAD);
    const float4* p1 = (const float4*)(sp + (size_t)(lane + 32) * DPAD);

    float wloss = 0.0f;
    for (int r = 0; r < 32; ++r) {
        const int i = rowbase + r;
        const float4* xr = (const float4*)(x + (size_t)(M + i) * D);

        float acc0 = 0.0f, acc1 = 0.0f;
        #pragma unroll 8
        for (int q = 0; q < D / 4; ++q) {
            float4 xv = xr[q];        // wave-broadcast load of the target row
            float4 a  = p0[q];
            float4 b  = p1[q];
            acc0 += signed_sq(xv.x - a.x) + signed_sq(xv.y - a.y)
                  + signed_sq(xv.z - a.z) + signed_sq(xv.w - a.w);
            acc1 += signed_sq(xv.x - b.x) + signed_sq(xv.y - b.y)
                  + signed_sq(xv.z - b.z) + signed_sq(xv.w - b.w);
        }
        float d0 = -acc0 * (1.0f / (float)D);   // distance for class  lane
        float d1 = -acc1 * (1.0f / (float)D);   // distance for class  lane+32

        // 64-wide log-softmax via wave32 reductions.
        float mx = fmaxf(d0, d1);
        #pragma unroll
        for (int s = 16; s > 0; s >>= 1) mx = fmaxf(mx, __shfl_xor(mx, s, 32));
        float e = __expf(d0 - mx) + __expf(d1 - mx);
        #pragma unroll
        for (int s = 16; s > 0; s >>= 1) e += __shfl_xor(e, s, 32);

        const int yc  = y[M + i];                      // wave-uniform
        float dsel    = (yc < 32) ? d0 : d1;
        float dtarget = __shfl(dsel, yc & 31, 32);
        wloss += -(dtarget - mx - __logf(e));          // same in every lane
    }

    if (lane == 0) wsum[wave] = wloss;
    __syncthreads();
    if (tid == 0) {
        float s = 0.0f;
        #pragma unroll
        for (int w = 0; w < 8; ++w) s += wsum[w];
        partials[blockIdx.x] = s;
    }
}

// ---------------------------------------------------------------------------
// Kernel 4: deterministic fixed-order final reduction -> scalar mean loss.
// ---------------------------------------------------------------------------
__global__ void k_final(const float* __restrict__ partials,
                        float* __restrict__ out) {
    if (threadIdx.x == 0 && blockIdx.x == 0) {
        float s = 0.0f;
        for (int i = 0; i < 64; ++i) s += partials[i];
        out[0] = s / (float)M;
    }
}

// ---------------------------------------------------------------------------
extern "C" void kernel_launch(void* const* d_in, const int* in_sizes, int n_in,
                              void* d_out, int out_size, void* d_ws, size_t ws_size,
                              hipStream_t stream) {
    (void)in_sizes; (void)n_in; (void)out_size; (void)ws_size;
    const float* x = (const float*)d_in[0];
    const int*   y = (const int*)d_in[1];     // JAX default x64-off => int32

    float* proto_sum = (float*)d_ws;          // NC*D floats
    float* counts    = proto_sum + NC * D;    // NC floats
    float* partials  = counts + NC;           // 64 floats

    k_counts<<<1, 256, 0, stream>>>(y, counts);
    k_proto <<<32, 32, 0, stream>>>(x, y, proto_sum);
    k_loss  <<<64, 256, 0, stream>>>(x, y, proto_sum, counts, partials);
    k_final <<<1, 32, 0, stream>>>(partials, (float*)d_out);
}